// Dense_SGCConv_22170621182124
// MI455X (gfx1250) — compile-verified
//
#include <hip/hip_runtime.h>
#include <hip/hip_bf16.h>

// Dense SGC conv for MI455X (gfx1250, wave32):
//   h    = x @ W + bias          (WMMA f32 16x16x4 matrix pipe, W in LDS)
//   CSR-ify edges by destination row (counting sort, int atomics only)
//   out  = pull-SpMM: one wave per node row, register accumulation,
//          fused deg-normalize + num_nodes mask, single write per row.
//
// B=8, N=10000, E=320000, C_IN=C_OUT=128 (C fixed to 128 by tiling).

typedef __attribute__((ext_vector_type(2))) float v2f;
typedef __attribute__((ext_vector_type(8))) float v8f;
typedef __attribute__((ext_vector_type(4))) int   v4i;

#define C_DIM 128
#define AS1 __attribute__((address_space(1)))
#define AS3 __attribute__((address_space(3)))

#if defined(__gfx1250__) &&                                     \
    __has_builtin(__builtin_amdgcn_global_load_async_to_lds_b128) && \
    __has_builtin(__builtin_amdgcn_s_wait_asynccnt)
#define USE_ASYNC_LDS 1
#else
#define USE_ASYNC_LDS 0
#endif

// ---------------------------------------------------------------------------
// Phase 1: h[row, :] = x[row, :] @ W + bias, rows = B*N.
// Block = 256 threads = 8 waves; 128-row strip per block.
// Each wave: 16 rows x 128 cols via 8 accumulators of v_wmma_f32_16x16x4_f32.
// W (128x128 f32 = 64KB) staged into LDS (async global->LDS when available).
// ---------------------------------------------------------------------------
__global__ __launch_bounds__(256) void gemm_bias_wmma(
    const float* __restrict__ X, const float* __restrict__ W,
    const float* __restrict__ bias, float* __restrict__ H, int n_rows) {
  __shared__ float sW[C_DIM * C_DIM];

#if USE_ASYNC_LDS
  {
    AS1 v4i* g = (AS1 v4i*)(const void*)W;   // 16B vectors in global
    AS3 v4i* l = (AS3 v4i*)(void*)sW;        // 16B vectors in LDS
    for (int i = threadIdx.x; i < (C_DIM * C_DIM) / 4; i += blockDim.x)
      __builtin_amdgcn_global_load_async_to_lds_b128(g + i, l + i, 0, 0);
    __builtin_amdgcn_s_wait_asynccnt(0);
  }
#else
  {
    const float4* src = (const float4*)W;
    float4* dst = (float4*)sW;
    for (int i = threadIdx.x; i < (C_DIM * C_DIM) / 4; i += blockDim.x)
      dst[i] = src[i];
  }
#endif
  __syncthreads();

  const int wave = threadIdx.x >> 5;
  const int lane = threadIdx.x & 31;
  const int half = lane >> 4;  // which 16-lane half
  const int l16  = lane & 15;

  const int row0 = blockIdx.x * 128 + wave * 16;  // uniform per wave
  if (row0 >= n_rows) return;

  if (row0 + 16 > n_rows) {
    // Scalar tail for ragged row counts (not hit for 80000 rows).
    for (int m = row0; m < n_rows; ++m) {
      const float* xr = X + (size_t)m * C_DIM;
      for (int col = lane; col < C_DIM; col += 32) {
        float acc = bias[col];
        for (int k = 0; k < C_DIM; ++k) acc += xr[k] * sW[k * C_DIM + col];
        H[(size_t)m * C_DIM + col] = acc;
      }
    }
    return;
  }

  // C/D 16x16 f32 layout: VGPR r -> (M = r + 8*half, N = l16). Init with bias.
  v8f acc[8];
#pragma unroll
  for (int t = 0; t < 8; ++t) {
    const float bt = bias[t * 16 + l16];
#pragma unroll
    for (int r = 0; r < 8; ++r) acc[t][r] = bt;
  }

  // A 16x4 f32 layout: lane half selects K pair {0,1} vs {2,3}; M = l16.
  const float* xrow = X + (size_t)(row0 + l16) * C_DIM;

  for (int kk = 0; kk < C_DIM / 4; ++kk) {
    const int kb = kk * 4 + 2 * half;        // even -> 8B aligned
    const v2f a = *(const v2f*)(xrow + kb);  // A[K=kb], A[K=kb+1]
#pragma unroll
    for (int t = 0; t < 8; ++t) {
      v2f b;  // B 4x16: row K striped on lanes
      b.x = sW[(kb + 0) * C_DIM + t * 16 + l16];
      b.y = sW[(kb + 1) * C_DIM + t * 16 + l16];
      acc[t] = __builtin_amdgcn_wmma_f32_16x16x4_f32(
          /*neg_a=*/false, a, /*neg_b=*/false, b,
          /*c_mod=*/(short)0, acc[t], /*reuse_a=*/false, /*reuse_b=*/false);
    }
  }

#pragma unroll
  for (int t = 0; t < 8; ++t) {
#pragma unroll
    for (int r = 0; r < 8; ++r) {
      const int m = row0 + r + 8 * half;
      H[(size_t)m * C_DIM + t * 16 + l16] = acc[t][r];
    }
  }
}

// ---------------------------------------------------------------------------
// Phase 2a: histogram of destination rows (int atomics; ~2.56M total).
// ---------------------------------------------------------------------------
__global__ __launch_bounds__(256) void count_rows(
    const int* __restrict__ erow, int* __restrict__ cnt, int N, int E,
    int total_edges) {
  const int t = blockIdx.x * blockDim.x + threadIdx.x;
  if (t >= total_edges) return;
  const int b = t / E;
  atomicAdd(&cnt[b * N + erow[t]], 1);
}

// ---------------------------------------------------------------------------
// Phase 2b: per-graph exclusive scan of cnt -> off (graph-local offsets).
// One 1024-thread block per graph; Hillis-Steele over 1024-chunks with carry.
// Writes both off[] and cursor[] (cursor consumed by the bucket scatter).
// ---------------------------------------------------------------------------
__global__ __launch_bounds__(1024) void scan_rows(
    const int* __restrict__ cnt, int* __restrict__ off,
    int* __restrict__ cursor, int N) {
  __shared__ int tmp[1024];
  __shared__ int carry;
  const int b = blockIdx.x;
  const int tid = threadIdx.x;
  if (tid == 0) carry = 0;
  __syncthreads();
  for (int base = 0; base < N; base += 1024) {
    const int i = base + tid;
    const int v = (i < N) ? cnt[b * N + i] : 0;
    tmp[tid] = v;
    __syncthreads();
    for (int d = 1; d < 1024; d <<= 1) {
      const int t = (tid >= d) ? tmp[tid - d] : 0;
      __syncthreads();
      tmp[tid] += t;
      __syncthreads();
    }
    const int excl = tmp[tid] - v;  // exclusive prefix within chunk
    const int c = carry;
    if (i < N) {
      off[b * N + i] = c + excl;
      cursor[b * N + i] = c + excl;
    }
    __syncthreads();
    if (tid == 1023) carry = c + tmp[1023];
    __syncthreads();
  }
}

// ---------------------------------------------------------------------------
// Phase 2c: scatter global edge indices into row buckets.
// ---------------------------------------------------------------------------
__global__ __launch_bounds__(256) void bucket_edges(
    const int* __restrict__ erow, int* __restrict__ cursor,
    int* __restrict__ sidx, int N, int E, int total_edges) {
  const int t = blockIdx.x * blockDim.x + threadIdx.x;
  if (t >= total_edges) return;
  const int b = t / E;
  const int pos = atomicAdd(&cursor[b * N + erow[t]], 1);  // graph-local slot
  sidx[(size_t)b * E + pos] = t;                           // global edge id
}

// ---------------------------------------------------------------------------
// Phase 3: pull-SpMM + normalize + mask. One wave per node row.
// Lane i owns channels [4i, 4i+4). Gathers h[b, col] (L2-resident:
// 5.12MB/graph << 192MB L2), accumulates in registers, writes once.
// ---------------------------------------------------------------------------
__global__ __launch_bounds__(256) void pull_spmm(
    const float* __restrict__ H, const int* __restrict__ ecol,
    const float* __restrict__ eval, const int* __restrict__ sidx,
    const int* __restrict__ off, const int* __restrict__ cnt,
    const int* __restrict__ num_nodes, float* __restrict__ out,
    int N, int E, int n_rows) {
  const int wave = blockIdx.x * (blockDim.x >> 5) + (threadIdx.x >> 5);
  const int lane = threadIdx.x & 31;
  if (wave >= n_rows) return;
  const int b = wave / N;
  const int n = wave - b * N;

  float4 acc = make_float4(0.f, 0.f, 0.f, 0.f);
  if (n < num_nodes[b]) {
    const int start = off[wave];
    const int cnt_n = cnt[wave];
    const int* bucket = sidx + (size_t)b * E + start;
    float deg = 0.f;
    for (int k = 0; k < cnt_n; ++k) {
      const int t = bucket[k];          // wave-uniform
      const int col = ecol[t];
      const float v = eval[t];
      const float4 m =
          ((const float4*)(H + ((size_t)(b * N + col)) * C_DIM))[lane];
      acc.x += v * m.x;
      acc.y += v * m.y;
      acc.z += v * m.z;
      acc.w += v * m.w;
      deg += v;
    }
    const float s = 1.0f / fmaxf(deg, 1.0f);
    acc.x *= s; acc.y *= s; acc.z *= s; acc.w *= s;
  }
  ((float4*)(out + (size_t)wave * C_DIM))[lane] = acc;
}

// ---------------------------------------------------------------------------
extern "C" void kernel_launch(void* const* d_in, const int* in_sizes, int n_in,
                              void* d_out, int out_size, void* d_ws,
                              size_t ws_size, hipStream_t stream) {
  const float* x    = (const float*)d_in[0];
  const float* w    = (const float*)d_in[1];
  const float* bias = (const float*)d_in[2];
  const int*   erow = (const int*)d_in[3];
  const int*   ecol = (const int*)d_in[4];
  const float* eval = (const float*)d_in[5];
  const int*   nn   = (const int*)d_in[6];

  const int B = in_sizes[6];
  const int N = in_sizes[0] / (B * C_DIM);
  const int E = in_sizes[3] / B;
  const int rows = B * N;
  const int total_edges = B * E;

  // Workspace layout.
  float* h    = (float*)d_ws;                       // rows * 128 f32
  int* cnt    = (int*)(h + (size_t)rows * C_DIM);   // rows i32
  int* off    = cnt + rows;                         // rows i32
  int* cursor = off + rows;                         // rows i32
  int* sidx   = cursor + rows;                      // B*E  i32

  (void)hipMemsetAsync(cnt, 0, (size_t)rows * sizeof(int), stream);

  // Phase 1: WMMA GEMM.
  gemm_bias_wmma<<<(rows + 127) / 128, 256, 0, stream>>>(x, w, bias, h, rows);

  // Phase 2: counting sort of edges by destination row.
  const int eblocks = (total_edges + 255) / 256;
  count_rows<<<eblocks, 256, 0, stream>>>(erow, cnt, N, E, total_edges);
  scan_rows<<<B, 1024, 0, stream>>>(cnt, off, cursor, N);
  bucket_edges<<<eblocks, 256, 0, stream>>>(erow, cursor, sidx, N, E,
                                            total_edges);

  // Phase 3: pull-SpMM with fused normalization + mask (one wave per row).
  const int pblocks = (rows + 7) / 8;  // 8 waves per 256-thread block
  pull_spmm<<<pblocks, 256, 0, stream>>>(h, ecol, eval, sidx, off, cnt, nn,
                                         (float*)d_out, N, E, rows);
}